// KNNSpaceRegularizer_82282983456988
// MI455X (gfx1250) — compile-verified
//
#include <hip/hip_runtime.h>

#define B_    8
#define N_    4096
#define D_    3
#define MAXK  16
#define NTILE (N_ / 16)

typedef __attribute__((ext_vector_type(2))) float v2f;
typedef __attribute__((ext_vector_type(8))) float v8f;

// One wave (32 threads) per block. Each block owns one batch b and one tile of
// 16 query rows [i0, i0+16). It sweeps all 256 column tiles of that batch,
// computing a 16x16 inner-product tile per step with V_WMMA_F32_16X16X4_F32
// (K=4, D=3 padded with zero), scores candidates by ||x_j||^2 - 2*x_i.x_j
// (same ordering as the true distance), and maintains a runtime-k nearest list
// per query in LDS. A branchless per-row tile-min fast-rejects tiles that
// cannot improve the current top-k (the common case once the list warms up).
__global__ __launch_bounds__(32) void knn_space_reg_kernel(
    const float* __restrict__ x,
    const float* __restrict__ preds,
    const float* __restrict__ kvec,
    float* __restrict__ out) {
  __shared__ float s_tile[16][17];     // transposed score tile (pad to 17 cols)
  __shared__ float s_bestd[16][MAXK];  // per-query k smallest scores
  __shared__ int   s_bestj[16][MAXK];  // per-query neighbor indices
  __shared__ int   s_k;

  const int lane = threadIdx.x;        // wave32
  const int b    = blockIdx.x / NTILE;
  const int i0   = (blockIdx.x % NTILE) * 16;

  // k = argmax(k_vector) + 1 (softmax is monotone; first-index tie break).
  if (lane == 0) {
    float bv = kvec[0];
    int   am = 0;
#pragma unroll
    for (int t = 1; t < MAXK; ++t) {
      float v = kvec[t];
      if (v > bv) { bv = v; am = t; }
    }
    s_k = am + 1;
  }
  if (lane < 16) {
#pragma unroll
    for (int t = 0; t < MAXK; ++t) {
      s_bestd[lane][t] = 3.0e38f;
      s_bestj[lane][t] = 0;
    }
  }
  __syncthreads();
  const int kk = s_k;

  const int    m     = lane & 15;      // row (for A) / col (for B) owned by lane
  const bool   hi    = lane >= 16;     // lanes 16-31 carry K=2,3 per ISA layout
  const size_t xbase = (size_t)b * N_ * 3;

  // A matrix: 16x4 f32, rows = queries i0..i0+15, K = {x,y,z,0}.
  // ISA layout: lanes 0-15 hold (K0,K1) of row M=lane; lanes 16-31 hold (K2,K3).
  const float* xm = x + xbase + (size_t)(i0 + m) * 3;
  const float  a0 = xm[0], a1 = xm[1], a2 = xm[2];
  v2f A;
  A.x = hi ? a2 : a0;
  A.y = hi ? 0.0f : a1;

  // Selection state (meaningful for lanes 0-15 only).
  float worst    = 3.0e38f;
  int   worstPos = 0;

  for (int jt = 0; jt < NTILE; ++jt) {
    const int j0 = jt * 16;

    // B matrix: 4x16 f32, cols = candidates j0..j0+15, K split like A.
    const float* xn = x + xbase + (size_t)(j0 + m) * 3;
    const float  c0 = xn[0], c1 = xn[1], c2 = xn[2];
    // WGP-scope prefetch of the next column tile (pull into near caches).
    if (jt + 1 < NTILE) __builtin_prefetch(xn + 48, 0, 3);

    v2f Bm;
    Bm.x = hi ? c2 : c0;
    Bm.y = hi ? 0.0f : c1;
    const float sq = c0 * c0 + c1 * c1 + c2 * c2;  // ||x_{j0+m}||^2

    // inner[m][n] = x_i . x_j for the 16x16 tile (one WMMA, K=4).
    v8f acc = {};
    acc = __builtin_amdgcn_wmma_f32_16x16x4_f32(
        /*neg_a=*/false, A, /*neg_b=*/false, Bm,
        /*c_mod=*/(short)0, acc, /*reuse_a=*/false, /*reuse_b=*/false);

    // C/D layout: lane holds column n = lane&15; VGPR v holds row v + 8*(lane>=16).
    // Score = ||x_j||^2 - 2*inner  (drop per-row constant ||x_i||^2: same order).
    const int rbase = hi ? 8 : 0;
#pragma unroll
    for (int v = 0; v < 8; ++v) {
      s_tile[rbase + v][m] = sq - 2.0f * acc[v];
    }
    __syncthreads();

    // Lanes 0-15: branchless row-min, then (rarely) run the insertion loop.
    if (lane < 16) {
      float d0, d1, d2, d3, d4, d5, d6, d7;
      float d8, d9, d10, d11, d12, d13, d14, d15;
      d0  = s_tile[lane][0];  d1  = s_tile[lane][1];
      d2  = s_tile[lane][2];  d3  = s_tile[lane][3];
      d4  = s_tile[lane][4];  d5  = s_tile[lane][5];
      d6  = s_tile[lane][6];  d7  = s_tile[lane][7];
      d8  = s_tile[lane][8];  d9  = s_tile[lane][9];
      d10 = s_tile[lane][10]; d11 = s_tile[lane][11];
      d12 = s_tile[lane][12]; d13 = s_tile[lane][13];
      d14 = s_tile[lane][14]; d15 = s_tile[lane][15];
      float tmin = fminf(
          fminf(fminf(fminf(d0, d1), fminf(d2, d3)),
                fminf(fminf(d4, d5), fminf(d6, d7))),
          fminf(fminf(fminf(d8, d9), fminf(d10, d11)),
                fminf(fminf(d12, d13), fminf(d14, d15))));

      if (tmin < worst) {
        // Slow path: at least one candidate improves the list.
        for (int n = 0; n < 16; ++n) {
          const float d = s_tile[lane][n];
          if (d < worst) {
            s_bestd[lane][worstPos] = d;
            s_bestj[lane][worstPos] = j0 + n;
            // Rescan the k active entries for the new worst.
            float w  = -3.0e38f;
            int   wp = 0;
            for (int t = 0; t < kk; ++t) {
              const float bd = s_bestd[lane][t];
              if (bd > w) { w = bd; wp = t; }
            }
            worst    = w;
            worstPos = wp;
          }
        }
      }
    }
    __syncthreads();
  }

  // Average preds over the k selected neighbors and write [B,N,3] output.
  if (lane < 16) {
    float sx = 0.0f, sy = 0.0f, sz = 0.0f;
    for (int t = 0; t < kk; ++t) {
      const int    j = s_bestj[lane][t];
      const float* p = preds + xbase + (size_t)j * 3;
      sx += p[0];
      sy += p[1];
      sz += p[2];
    }
    const float inv = 1.0f / (float)kk;
    float* o = out + xbase + (size_t)(i0 + lane) * 3;
    o[0] = sx * inv;
    o[1] = sy * inv;
    o[2] = sz * inv;
  }
}

extern "C" void kernel_launch(void* const* d_in, const int* in_sizes, int n_in,
                              void* d_out, int out_size, void* d_ws, size_t ws_size,
                              hipStream_t stream) {
  (void)in_sizes; (void)n_in; (void)out_size; (void)d_ws; (void)ws_size;
  const float* x     = (const float*)d_in[0];
  const float* preds = (const float*)d_in[1];
  const float* kvec  = (const float*)d_in[2];
  float*       out   = (float*)d_out;

  const int nblocks = B_ * NTILE;  // 8 * 256 = 2048 blocks, one wave32 each
  knn_space_reg_kernel<<<nblocks, 32, 0, stream>>>(x, preds, kvec, out);
}